// HIST_67585605370195
// MI455X (gfx1250) — compile-verified
//
#include <hip/hip_runtime.h>
#include <math.h>

#define NN 4096
#define CC 512
#define HH 128
#define TT 64
#define DD 20
#define KTOP 10

typedef __attribute__((ext_vector_type(2))) float v2f;
typedef __attribute__((ext_vector_type(8))) float v8f;

// ---------------- WMMA helpers (V_WMMA_F32_16X16X4_F32, exact fp32) ----------------

static __device__ __forceinline__ v8f wmma4(v2f a, v2f b, v8f c) {
  // 8 args: (neg_a, A, neg_b, B, c_mod, C, reuse_a, reuse_b)
  return __builtin_amdgcn_wmma_f32_16x16x4_f32(false, a, false, b, (short)0, c, false, false);
}

// A fragment: 16x4 tile at (row0, k0); element A[r][k] at A + r*ak + k*an
// lanes 0-15 -> M=lane, K={k0,k0+1}; lanes 16-31 -> M=lane-16, K={k0+2,k0+3}
// Contiguous (an==1) case: single 8B-aligned b64 load.
static __device__ __forceinline__ v2f frag_a_c(const float* __restrict__ A,
                                               long row0, long k0, long ak, int lane) {
  int m  = lane & 15;
  int ko = (lane < 16) ? 0 : 2;
  return *(const v2f*)(A + (row0 + m) * ak + (k0 + ko));
}
static __device__ __forceinline__ v2f frag_a_s(const float* __restrict__ A,
                                               long row0, long k0, long ak, long an, int lane) {
  int m  = lane & 15;
  int ko = (lane < 16) ? 0 : 2;
  const float* p = A + (row0 + m) * ak + (k0 + ko) * an;
  v2f r; r.x = p[0]; r.y = p[an];
  return r;
}

// B fragment: 4x16 tile at (k0, col0); element B[k][n] at B + k*bk + n*bn
static __device__ __forceinline__ v2f frag_b_c(const float* __restrict__ B,
                                               long k0, long col0, long bn, int lane) {
  int n  = lane & 15;
  int ko = (lane < 16) ? 0 : 2;
  return *(const v2f*)(B + (k0 + ko) + (col0 + n) * bn);
}
static __device__ __forceinline__ v2f frag_b_s(const float* __restrict__ B,
                                               long k0, long col0, long bk, long bn, int lane) {
  int n  = lane & 15;
  int ko = (lane < 16) ? 0 : 2;
  const float* p = B + (k0 + ko) * bk + (col0 + n) * bn;
  v2f r; r.x = p[0]; r.y = p[bk];
  return r;
}

// D tile 16x16: element r of v8f is (row0 + (lane<16?0:8) + r, col0 + (lane&15))

// ---------------- generic WMMA GEMM ----------------
// C[M,Ncol] = act( scale( A(ak,an) x B(bk,bn) ) + bias )
// block = 256 threads (8 waves); wave tile 32x32 (4 accumulators; 4 b64 loads /
// 4 wmma per k-step in the contiguous case); block tile 64x128;
// grid (M/64, Ncol/128). M mult of 64, Ncol mult of 128, Kdim mult of 4.
// AC/BC: compile-time contiguity of the A (an==1) / B (bk==1) fragments.
template <bool AC, bool BC>
__global__ void wmma_gemm(const float* __restrict__ A, long ak, long an,
                          const float* __restrict__ B, long bk, long bn,
                          float* __restrict__ Cout, long ldc,
                          int M, int Ncol, int Kdim,
                          const float* __restrict__ bias, int act,
                          const float* __restrict__ rscale, const float* __restrict__ cscale) {
  int lane = threadIdx.x & 31;
  int wave = threadIdx.x >> 5;
  long row0 = (long)blockIdx.x * 64 + (wave & 1) * 32;
  long col0 = (long)blockIdx.y * 128 + (wave >> 1) * 32;
  v8f acc[4];
#pragma unroll
  for (int i = 0; i < 4; i++) acc[i] = (v8f){0,0,0,0,0,0,0,0};
  for (int k = 0; k < Kdim; k += 4) {
    v2f a0 = AC ? frag_a_c(A, row0,      k, ak, lane) : frag_a_s(A, row0,      k, ak, an, lane);
    v2f a1 = AC ? frag_a_c(A, row0 + 16, k, ak, lane) : frag_a_s(A, row0 + 16, k, ak, an, lane);
    v2f b0 = BC ? frag_b_c(B, k, col0,      bn, lane) : frag_b_s(B, k, col0,      bk, bn, lane);
    v2f b1 = BC ? frag_b_c(B, k, col0 + 16, bn, lane) : frag_b_s(B, k, col0 + 16, bk, bn, lane);
    acc[0] = wmma4(a0, b0, acc[0]);
    acc[1] = wmma4(a0, b1, acc[1]);
    acc[2] = wmma4(a1, b0, acc[2]);
    acc[3] = wmma4(a1, b1, acc[3]);
  }
  int n  = lane & 15;
  int mb = (lane < 16) ? 0 : 8;
#pragma unroll
  for (int i = 0; i < 2; i++)
#pragma unroll
    for (int j = 0; j < 2; j++) {
      v8f a = acc[i * 2 + j];
      long col = col0 + j * 16 + n;
      float bv = bias   ? bias[col]   : 0.f;
      float cs = cscale ? cscale[col] : 1.f;
#pragma unroll
      for (int r = 0; r < 8; r++) {
        long row = row0 + i * 16 + mb + r;
        float v = a[r];
        if (rscale) v = v / fmaxf(rscale[row] * cs, 1e-12f);
        v += bv;
        if (act) v = (v > 0.f) ? v : 0.01f * v;
        Cout[row * ldc + col] = v;
      }
    }
}

// ---------------- specialized: hiddenC[C,H] = s2c.T @ x_hidden ----------------
// s2c[n,c] = cm[n,c]*mv[n] / (colsum[c]*cm[n,c] + 1)
__global__ void wmma_gemm_s2c(const float* __restrict__ cm, const float* __restrict__ mv,
                              const float* __restrict__ colsum, const float* __restrict__ xh,
                              float* __restrict__ Cout) {
  int lane = threadIdx.x & 31;
  int wave = threadIdx.x >> 5;
  long row0 = (long)blockIdx.x * 64 + (wave & 3) * 16;   // concept index c
  long col0 = (long)blockIdx.y * 64 + (wave >> 2) * 32;  // h index
  int m  = lane & 15;
  int ko = (lane < 16) ? 0 : 2;
  long c = row0 + m;
  float csv = colsum[c];
  v8f acc0 = {0,0,0,0,0,0,0,0};
  v8f acc1 = {0,0,0,0,0,0,0,0};
  for (int k = 0; k < NN; k += 4) {
    float m0 = cm[(long)(k + ko) * CC + c];
    float m1 = cm[(long)(k + ko + 1) * CC + c];
    v2f a;
    a.x = m0 * mv[k + ko]     / (csv * m0 + 1.f);
    a.y = m1 * mv[k + ko + 1] / (csv * m1 + 1.f);
    v2f b0 = frag_b_s(xh, k, col0,      HH, 1, lane);
    v2f b1 = frag_b_s(xh, k, col0 + 16, HH, 1, lane);
    acc0 = wmma4(a, b0, acc0);
    acc1 = wmma4(a, b1, acc1);
  }
  int mb = (lane < 16) ? 0 : 8;
#pragma unroll
  for (int tile = 0; tile < 2; tile++) {
    v8f acc = tile ? acc1 : acc0;
    long col = col0 + tile * 16 + m;
#pragma unroll
    for (int r = 0; r < 8; r++) Cout[(row0 + mb + r) * HH + col] = acc[r];
  }
}

// ---------------- fused GRU step (one layer, one timestep) ----------------
// block = 128 threads (4 waves); wave -> 16 rows x 32 h-cols; grid (NN/64, HH/32)
// gates: r,z,n segments of [N,384]; kept entirely in 12 v8f accumulators.
// XC: compile-time contiguity of the x-input fragments (layer1 true, layer0 false).
template <bool XC>
__global__ void gru_step(const float* __restrict__ Abase, long ak, long an, int in_dim,
                         const float* __restrict__ wih, const float* __restrict__ whh,
                         const float* __restrict__ bih, const float* __restrict__ bhh,
                         const float* __restrict__ h_in, float* __restrict__ h_out) {
  int lane = threadIdx.x & 31;
  int wave = threadIdx.x >> 5;
  long row0 = (long)blockIdx.x * 64 + wave * 16;
  int  c0   = blockIdx.y * 32;

  v8f gh[6], gx[6];
#pragma unroll
  for (int i = 0; i < 6; i++) { gh[i] = (v8f){0,0,0,0,0,0,0,0}; gx[i] = (v8f){0,0,0,0,0,0,0,0}; }

  for (int k = 0; k < HH; k += 4) {              // gh = h_in @ whh.T
    v2f a = frag_a_c(h_in, row0, k, HH, lane);
#pragma unroll
    for (int s = 0; s < 3; s++)
#pragma unroll
      for (int t2 = 0; t2 < 2; t2++) {
        v2f b = frag_b_c(whh, k, s * HH + c0 + t2 * 16, HH, lane);
        gh[s * 2 + t2] = wmma4(a, b, gh[s * 2 + t2]);
      }
  }
  for (int k = 0; k < in_dim; k += 4) {          // gx = xt @ wih.T
    v2f a = XC ? frag_a_c(Abase, row0, k, ak, lane) : frag_a_s(Abase, row0, k, ak, an, lane);
#pragma unroll
    for (int s = 0; s < 3; s++)
#pragma unroll
      for (int t2 = 0; t2 < 2; t2++) {
        v2f b = frag_b_c(wih, k, s * HH + c0 + t2 * 16, in_dim, lane);
        gx[s * 2 + t2] = wmma4(a, b, gx[s * 2 + t2]);
      }
  }
  int nidx = lane & 15;
  int mb   = (lane < 16) ? 0 : 8;
#pragma unroll
  for (int t2 = 0; t2 < 2; t2++) {
    int c = c0 + t2 * 16 + nidx;
    float bxr = bih[c], bxz = bih[HH + c], bxn = bih[2 * HH + c];
    float bhr = bhh[c], bhz = bhh[HH + c], bhn = bhh[2 * HH + c];
#pragma unroll
    for (int r = 0; r < 8; r++) {
      long row = row0 + mb + r;
      float grv = gx[0 + t2][r] + bxr + gh[0 + t2][r] + bhr;
      float gzv = gx[2 + t2][r] + bxz + gh[2 + t2][r] + bhz;
      float rg = 1.f / (1.f + expf(-grv));
      float zg = 1.f / (1.f + expf(-gzv));
      float ng = tanhf(gx[4 + t2][r] + bxn + rg * (gh[4 + t2][r] + bhn));
      float ho = h_in[row * HH + c];
      h_out[row * HH + c] = (1.f - zg) * ng + zg * ho;
    }
  }
}

// ---------------- reductions / softmaxes / elementwise ----------------

__global__ void zero_kernel(float* p, long n) {
  long i = (long)blockIdx.x * blockDim.x + threadIdx.x;
  long st = (long)gridDim.x * blockDim.x;
  for (; i < n; i += st) p[i] = 0.f;
}

__global__ void colsum_s2c_kernel(const float* __restrict__ cm, const float* __restrict__ mv,
                                  float* __restrict__ colsum) {
  __shared__ float red[256];
  int c = blockIdx.x, tid = threadIdx.x;
  float s = 0.f;
  for (int n = tid; n < NN; n += 256) s += cm[(long)n * CC + c] * mv[n];
  red[tid] = s; __syncthreads();
  for (int st = 128; st > 0; st >>= 1) { if (tid < st) red[tid] += red[tid + st]; __syncthreads(); }
  if (tid == 0) colsum[c] = red[0];
}

// mode 0: out1=sqrt(sum v^2), out2(optional)=sum v^2 ; mode 1: out1 = (sum v != 0) ? 1 : 0
__global__ void row_reduce_kernel(const float* __restrict__ mat, int cols,
                                  float* __restrict__ out1, float* __restrict__ out2, int mode) {
  __shared__ float red[128];
  int r = blockIdx.x, tid = threadIdx.x;
  float s = 0.f;
  for (int c = tid; c < cols; c += 128) {
    float v = mat[(long)r * cols + c];
    s += (mode == 0) ? v * v : v;
  }
  red[tid] = s; __syncthreads();
  for (int st = 64; st > 0; st >>= 1) { if (tid < st) red[tid] += red[tid + st]; __syncthreads(); }
  if (tid == 0) {
    float t = red[0];
    if (mode == 0) { out1[r] = sqrtf(t); if (out2) out2[r] = t; }
    else out1[r] = (t != 0.f) ? 1.f : 0.f;
  }
}

__global__ void col_softmax_kernel(float* mat, int rows, int cols) {
  __shared__ float red[256];
  __shared__ float sM, sS;
  int c = blockIdx.x, tid = threadIdx.x;
  float m = -3.4e38f;
  for (int n = tid; n < rows; n += 256) m = fmaxf(m, mat[(long)n * cols + c]);
  red[tid] = m; __syncthreads();
  for (int st = 128; st > 0; st >>= 1) { if (tid < st) red[tid] = fmaxf(red[tid], red[tid + st]); __syncthreads(); }
  if (tid == 0) sM = red[0];
  __syncthreads();
  float M = sM;
  float su = 0.f;
  for (int n = tid; n < rows; n += 256) su += expf(mat[(long)n * cols + c] - M);
  red[tid] = su; __syncthreads();
  for (int st = 128; st > 0; st >>= 1) { if (tid < st) red[tid] += red[tid + st]; __syncthreads(); }
  if (tid == 0) sS = red[0];
  __syncthreads();
  float inv = 1.f / sS;
  for (int n = tid; n < rows; n += 256) mat[(long)n * cols + c] = expf(mat[(long)n * cols + c] - M) * inv;
}

// row softmax with column validity mask; p = softmax(where(v,s,-inf)) * v  (in place)
__global__ void row_softmax_mask_kernel(float* mat, const float* __restrict__ valid, int W) {
  __shared__ float red[256];
  __shared__ float sM, sS;
  int r = blockIdx.x, tid = threadIdx.x;
  long base = (long)r * W;
  float m = -3.4e38f;
  for (int j = tid; j < W; j += 256) if (valid[j] != 0.f) m = fmaxf(m, mat[base + j]);
  red[tid] = m; __syncthreads();
  for (int st = 128; st > 0; st >>= 1) { if (tid < st) red[tid] = fmaxf(red[tid], red[tid + st]); __syncthreads(); }
  if (tid == 0) sM = red[0];
  __syncthreads();
  float M = sM;
  float su = 0.f;
  for (int j = tid; j < W; j += 256) if (valid[j] != 0.f) su += expf(mat[base + j] - M);
  red[tid] = su; __syncthreads();
  for (int st = 128; st > 0; st >>= 1) { if (tid < st) red[tid] += red[tid + st]; __syncthreads(); }
  if (tid == 0) sS = red[0];
  __syncthreads();
  float inv = (sS > 0.f) ? 1.f / sS : 0.f;
  for (int j = tid; j < W; j += 256) {
    float v = (valid[j] != 0.f) ? expf(mat[base + j] - M) * inv : 0.f;
    mat[base + j] = v;
  }
}

// ---------------- top-K + scatter ----------------

static __device__ __forceinline__ void topk_insert(float* v, int* id, float nv, int ni) {
  if (nv <= v[KTOP - 1]) return;
  int pos = KTOP - 1;
  while (pos > 0 && v[pos - 1] < nv) { v[pos] = v[pos - 1]; id[pos] = id[pos - 1]; pos--; }
  v[pos] = nv; id[pos] = ni;
}

__global__ void topk_kernel(const float* __restrict__ sim,
                            float* __restrict__ topv, int* __restrict__ topi,
                            float* __restrict__ colsum2) {
  __shared__ float sv[256 * KTOP];
  __shared__ int   si[256 * KTOP];
  int i = blockIdx.x, tid = threadIdx.x;
  float lv[KTOP]; int li[KTOP];
#pragma unroll
  for (int k = 0; k < KTOP; k++) { lv[k] = -3.4e38f; li[k] = -1; }
  for (int j = tid; j < NN; j += 256) {
    float v = (j == i) ? 0.f : sim[(long)i * NN + j];   // diag zeroed before top_k
    topk_insert(lv, li, v, j);
  }
  for (int k = 0; k < KTOP; k++) { sv[tid * KTOP + k] = lv[k]; si[tid * KTOP + k] = li[k]; }
  __syncthreads();
  for (int st = 128; st > 0; st >>= 1) {
    if (tid < st) {
      float* dv = &sv[tid * KTOP]; int* di = &si[tid * KTOP];
      float* ev = &sv[(tid + st) * KTOP]; int* ei = &si[(tid + st) * KTOP];
      for (int e = 0; e < KTOP; e++) topk_insert(dv, di, ev[e], ei[e]);
    }
    __syncthreads();
  }
  if (tid < KTOP) {
    float v = sv[tid]; int j = si[tid];
    topv[(long)i * KTOP + tid] = v;
    topi[(long)i * KTOP + tid] = j;
    if (j >= 0) atomicAdd(&colsum2[j], v);
  }
}

__global__ void scatter_hidden_kernel(const float* __restrict__ topv, const int* __restrict__ topi,
                                      const float* __restrict__ hshared, float* __restrict__ hiddenH) {
  int i = blockIdx.x, h = threadIdx.x;
  float xv = hshared[(long)i * HH + h];
  for (int k = 0; k < KTOP; k++) {
    float v = topv[(long)i * KTOP + k];
    int j = topi[(long)i * KTOP + k];
    if (j >= 0) atomicAdd(&hiddenH[(long)j * HH + h], v * xv);
  }
}

__global__ void diag_add_kernel(const float* __restrict__ colsum2, const float* __restrict__ hsq,
                                const float* __restrict__ hshared, float* __restrict__ hiddenH) {
  int j = blockIdx.x, h = threadIdx.x;
  if (colsum2[j] != 0.f) {
    float d = hsq[j] / fmaxf(hsq[j], 1e-12f);   // diagonal cosine value
    hiddenH[(long)j * HH + h] += d * hshared[(long)j * HH + h];
  }
}

__global__ void sub_kernel(const float* a, const float* b, float* o, long n) {
  long i = (long)blockIdx.x * blockDim.x + threadIdx.x;
  long st = (long)gridDim.x * blockDim.x;
  for (; i < n; i += st) o[i] = a[i] - b[i];
}

__global__ void sub3_kernel(const float* a, const float* b, const float* c, float* o, long n) {
  long i = (long)blockIdx.x * blockDim.x + threadIdx.x;
  long st = (long)gridDim.x * blockDim.x;
  for (; i < n; i += st) o[i] = a[i] - b[i] - c[i];
}

__global__ void head_kernel(const float* __restrict__ a, const float* __restrict__ b,
                            const float* __restrict__ c, const float* __restrict__ wout,
                            const float* __restrict__ bout, float* __restrict__ out) {
  __shared__ float red[128];
  int n = blockIdx.x, tid = threadIdx.x;
  long idx = (long)n * HH + tid;
  red[tid] = (a[idx] + b[idx] + c[idx]) * wout[tid];
  __syncthreads();
  for (int st = 64; st > 0; st >>= 1) { if (tid < st) red[tid] += red[tid + st]; __syncthreads(); }
  if (tid == 0) out[n] = red[0] + bout[0];
}

// ---------------- host launcher ----------------

extern "C" void kernel_launch(void* const* d_in, const int* in_sizes, int n_in,
                              void* d_out, int out_size, void* d_ws, size_t ws_size,
                              hipStream_t stream) {
  (void)in_sizes; (void)n_in; (void)out_size; (void)ws_size;
  const float* x    = (const float*)d_in[0];
  const float* cm   = (const float*)d_in[1];
  const float* mv   = (const float*)d_in[2];
  const float* wih0 = (const float*)d_in[3];
  const float* whh0 = (const float*)d_in[4];
  const float* bih0 = (const float*)d_in[5];
  const float* bhh0 = (const float*)d_in[6];
  const float* wih1 = (const float*)d_in[7];
  const float* whh1 = (const float*)d_in[8];
  const float* bih1 = (const float*)d_in[9];
  const float* bhh1 = (const float*)d_in[10];
  const float* w_ps = (const float*)d_in[11];      const float* b_ps = (const float*)d_in[12];
  const float* w_hs = (const float*)d_in[13];      const float* b_hs = (const float*)d_in[14];
  const float* w_ps_fore = (const float*)d_in[15]; const float* b_ps_fore = (const float*)d_in[16];
  const float* w_hs_fore = (const float*)d_in[17]; const float* b_hs_fore = (const float*)d_in[18];
  const float* w_ps_back = (const float*)d_in[19]; const float* b_ps_back = (const float*)d_in[20];
  const float* w_hs_back = (const float*)d_in[21]; const float* b_hs_back = (const float*)d_in[22];
  const float* w_indi = (const float*)d_in[23];    const float* b_indi = (const float*)d_in[24];
  const float* w_out = (const float*)d_in[25];     const float* b_out = (const float*)d_in[26];
  float* out = (float*)d_out;

  char* base = (char*)d_ws;
  size_t off = 0;
  auto alloc = [&](size_t nbytes) -> void* {
    void* q = base + off;
    off += (nbytes + 255) & ~(size_t)255;
    return q;
  };
  const size_t NH = (size_t)NN * HH * 4;
  float* h0a = (float*)alloc(NH);
  float* h0b = (float*)alloc(NH);
  float* h1a = (float*)alloc(NH);
  float* h1b = (float*)alloc(NH);
  float* colsumC = (float*)alloc(CC * 4);
  float* hiddenC = (float*)alloc((size_t)CC * HH * 4);
  float* v1 = (float*)alloc(CC * 4);
  float* probsNC = (float*)alloc((size_t)NN * CC * 4);
  float* hidden2 = (float*)alloc((size_t)CC * HH * 4);
  float* nx  = (float*)alloc(NN * 4);
  float* nyC = (float*)alloc(CC * 4);
  float* ppre    = (float*)alloc(NH);
  float* pshared = (float*)alloc(NH);
  float* pback   = (float*)alloc(NH);
  float* outps   = (float*)alloc(NH);
  float* hshared = (float*)alloc(NH);
  float* hn  = (float*)alloc(NN * 4);
  float* hsq = (float*)alloc(NN * 4);
  float* simNN = (float*)alloc((size_t)NN * NN * 4);
  float* topv = (float*)alloc((size_t)NN * KTOP * 4);
  int*   topi = (int*)alloc((size_t)NN * KTOP * 4);
  float* colsum2 = (float*)alloc(NN * 4);
  float* hiddenH = (float*)alloc(NH);
  float* nh2 = (float*)alloc(NN * 4);
  float* v2  = (float*)alloc(NN * 4);
  float* hsi    = (float*)alloc(NH);
  float* hsi2   = (float*)alloc(NH);
  float* hback  = (float*)alloc(NH);
  float* ouths  = (float*)alloc(NH);
  float* indi   = (float*)alloc(NH);
  float* outindi = (float*)alloc(NH);

  // static dispatch on fragment contiguity (an==1 / bk==1)
  auto gemm = [&](const float* A, long ak, long an, const float* B, long bk, long bn,
                  float* C, long ldc, int M, int Ncl, int Kd,
                  const float* bias, int act, const float* rs, const float* cs) {
    dim3 g(M / 64, Ncl / 128);
    if (an == 1 && bk == 1)
      wmma_gemm<true, true><<<g, 256, 0, stream>>>(A, ak, an, B, bk, bn, C, ldc, M, Ncl, Kd, bias, act, rs, cs);
    else if (an == 1)
      wmma_gemm<true, false><<<g, 256, 0, stream>>>(A, ak, an, B, bk, bn, C, ldc, M, Ncl, Kd, bias, act, rs, cs);
    else if (bk == 1)
      wmma_gemm<false, true><<<g, 256, 0, stream>>>(A, ak, an, B, bk, bn, C, ldc, M, Ncl, Kd, bias, act, rs, cs);
    else
      wmma_gemm<false, false><<<g, 256, 0, stream>>>(A, ak, an, B, bk, bn, C, ldc, M, Ncl, Kd, bias, act, rs, cs);
  };

  // ---- GRU (2 layers fused per timestep, ping-pong h buffers) ----
  zero_kernel<<<1024, 256, 0, stream>>>(h0a, (long)NN * HH);
  zero_kernel<<<1024, 256, 0, stream>>>(h1a, (long)NN * HH);
  float *h0_in = h0a, *h0_out = h0b, *h1_in = h1a, *h1_out = h1b;
  dim3 ggru(NN / 64, HH / 32);
  for (int t = 0; t < TT; t++) {
    // layer0: xt[n][d] = x[n*1280 + d*64 + t]  (strided x-fragments)
    gru_step<false><<<ggru, 128, 0, stream>>>(x + t, (long)DD * TT, (long)TT, DD,
                                              wih0, whh0, bih0, bhh0, h0_in, h0_out);
    // layer1: contiguous x-fragments
    gru_step<true><<<ggru, 128, 0, stream>>>(h0_out, (long)HH, 1L, HH,
                                             wih1, whh1, bih1, bhh1, h1_in, h1_out);
    float* tmp = h0_in; h0_in = h0_out; h0_out = tmp;
    tmp = h1_in; h1_in = h1_out; h1_out = tmp;
  }
  const float* xh = h1_in;  // x_hidden [N,H]

  // ---- predefined-concept branch ----
  colsum_s2c_kernel<<<CC, 256, 0, stream>>>(cm, mv, colsumC);
  wmma_gemm_s2c<<<dim3(CC / 64, HH / 64), 256, 0, stream>>>(cm, mv, colsumC, xh, hiddenC);
  row_reduce_kernel<<<CC, 128, 0, stream>>>(hiddenC, HH, v1, nullptr, 1);
  // logits [N,C] = xh @ hiddenC.T
  gemm(xh, HH, 1, hiddenC, 1, HH, probsNC, CC, NN, CC, HH, nullptr, 0, nullptr, nullptr);
  col_softmax_kernel<<<CC, 256, 0, stream>>>(probsNC, NN, CC);          // s2c2
  // hidden2 [C,H] = s2c2.T @ xh
  gemm(probsNC, 1, CC, xh, HH, 1, hidden2, HH, CC, HH, NN, nullptr, 0, nullptr, nullptr);
  row_reduce_kernel<<<NN, 128, 0, stream>>>(xh, HH, nx, nullptr, 0);
  row_reduce_kernel<<<CC, 128, 0, stream>>>(hidden2, HH, nyC, nullptr, 0);
  // c2s [N,C] = cos(xh, hidden2)  (reuse probsNC)
  gemm(xh, HH, 1, hidden2, 1, HH, probsNC, CC, NN, CC, HH, nullptr, 0, nx, nyC);
  row_softmax_mask_kernel<<<NN, 256, 0, stream>>>(probsNC, v1, CC);
  // p_shared = (c2s @ hidden2) @ w_ps.T + b_ps
  gemm(probsNC, CC, 1, hidden2, HH, 1, ppre, HH, NN, HH, CC, nullptr, 0, nullptr, nullptr);
  gemm(ppre, HH, 1, w_ps, 1, HH, pshared, HH, NN, HH, HH, b_ps, 0, nullptr, nullptr);
  gemm(pshared, HH, 1, w_ps_back, 1, HH, pback, HH, NN, HH, HH, b_ps_back, 0, nullptr, nullptr);
  gemm(pshared, HH, 1, w_ps_fore, 1, HH, outps, HH, NN, HH, HH, b_ps_fore, 1, nullptr, nullptr);

  // ---- hidden-concept branch ----
  sub_kernel<<<1024, 256, 0, stream>>>(xh, pback, hshared, (long)NN * HH);
  row_reduce_kernel<<<NN, 128, 0, stream>>>(hshared, HH, hn, hsq, 0);
  // sim [N,N] = cos(hshared, hshared)
  gemm(hshared, HH, 1, hshared, 1, HH, simNN, NN, NN, NN, HH, nullptr, 0, hn, hn);
  zero_kernel<<<64, 256, 0, stream>>>(colsum2, NN);
  zero_kernel<<<1024, 256, 0, stream>>>(hiddenH, (long)NN * HH);
  topk_kernel<<<NN, 256, 0, stream>>>(simNN, topv, topi, colsum2);
  scatter_hidden_kernel<<<NN, HH, 0, stream>>>(topv, topi, hshared, hiddenH);
  diag_add_kernel<<<NN, HH, 0, stream>>>(colsum2, hsq, hshared, hiddenH);
  row_reduce_kernel<<<NN, 128, 0, stream>>>(hiddenH, HH, v2, nullptr, 1);
  row_reduce_kernel<<<NN, 128, 0, stream>>>(hiddenH, HH, nh2, nullptr, 0);
  // hc2s [N,N] = cos(hshared, hiddenH)  (reuse simNN)
  gemm(hshared, HH, 1, hiddenH, 1, HH, simNN, NN, NN, NN, HH, nullptr, 0, hn, nh2);
  row_softmax_mask_kernel<<<NN, 256, 0, stream>>>(simNN, v2, NN);
  // h_si = (hc2s @ hiddenH) @ w_hs.T + b_hs
  gemm(simNN, NN, 1, hiddenH, HH, 1, hsi, HH, NN, HH, NN, nullptr, 0, nullptr, nullptr);
  gemm(hsi, HH, 1, w_hs, 1, HH, hsi2, HH, NN, HH, HH, b_hs, 0, nullptr, nullptr);
  gemm(hsi2, HH, 1, w_hs_back, 1, HH, hback, HH, NN, HH, HH, b_hs_back, 0, nullptr, nullptr);
  gemm(hsi2, HH, 1, w_hs_fore, 1, HH, ouths, HH, NN, HH, HH, b_hs_fore, 1, nullptr, nullptr);

  // ---- individual branch + head ----
  sub3_kernel<<<1024, 256, 0, stream>>>(xh, pback, hback, indi, (long)NN * HH);
  gemm(indi, HH, 1, w_indi, 1, HH, outindi, HH, NN, HH, HH, b_indi, 1, nullptr, nullptr);
  head_kernel<<<NN, HH, 0, stream>>>(outps, ouths, outindi, w_out, b_out, out);
}